// CuttingDetectionLoss_36704790511629
// MI455X (gfx1250) — compile-verified
//
#include <hip/hip_runtime.h>
#include <hip/hip_bf16.h>

#define NBLOCKS  2048
#define NTHREADS 256
#define NWAVES   (NTHREADS / 32)

typedef __attribute__((ext_vector_type(2))) float v2f;
typedef __attribute__((ext_vector_type(8))) float v8f;

// ---------------------------------------------------------------------------
// Wave32 reduction using V_WMMA_F32_16X16X4_F32.
// A = ones(16x4). B: VGPR0 holds rows K=0 (lanes 0-15) / K=2 (lanes 16-31),
// VGPR1 holds rows K=1 / K=3 and is set to 0. Since A is all-ones, every row
// of D equals the column sums, so d[0] in lane L = v[L%16] + v[(L%16)+16]
// (a correct fp32 32->16 fold). Finish 16->1 with xor shuffles.
// EXEC is all-ones at every call site (uniform control flow, full blocks).
// ---------------------------------------------------------------------------
__device__ __forceinline__ float wave_reduce_wmma(float v) {
    v2f a; a[0] = 1.0f; a[1] = 1.0f;
    v2f b; b[0] = v;    b[1] = 0.0f;
    v8f c = {};
    v8f d = __builtin_amdgcn_wmma_f32_16x16x4_f32(
        /*neg_a=*/false, a, /*neg_b=*/false, b,
        /*c_mod=*/(short)0, c, /*reuse_a=*/false, /*reuse_b=*/false);
    float s = d[0];
    s += __shfl_xor(s, 8, 32);
    s += __shfl_xor(s, 4, 32);
    s += __shfl_xor(s, 2, 32);
    s += __shfl_xor(s, 1, 32);
    return s;
}

__device__ __forceinline__ void block_reduce4(float a0, float a1, float a2, float a3,
                                              float* __restrict__ dst) {
    __shared__ float lds[NWAVES][4];
    a0 = wave_reduce_wmma(a0);
    a1 = wave_reduce_wmma(a1);
    a2 = wave_reduce_wmma(a2);
    a3 = wave_reduce_wmma(a3);
    const int lane = threadIdx.x & 31;
    const int wave = threadIdx.x >> 5;
    if (lane == 0) {
        lds[wave][0] = a0; lds[wave][1] = a1; lds[wave][2] = a2; lds[wave][3] = a3;
    }
    __syncthreads();
    if (threadIdx.x == 0) {
        float s0 = 0.f, s1 = 0.f, s2 = 0.f, s3 = 0.f;
#pragma unroll
        for (int w = 0; w < NWAVES; ++w) {
            s0 += lds[w][0]; s1 += lds[w][1]; s2 += lds[w][2]; s3 += lds[w][3];
        }
        dst[0] = s0; dst[1] = s1; dst[2] = s2; dst[3] = s3;
    }
}

// _validate_and_fix from the reference (branchless; per-lane cndmask).
__device__ __forceinline__ float4 fix_box(float4 b) {
    const bool inv = (b.x > b.z) || (b.y > b.w);
    float x1 = inv ? fmaxf(b.x, 0.f) : b.x;
    float y1 = inv ? fmaxf(b.y, 0.f) : b.y;
    float x2 = inv ? fmaxf(b.z, 0.f) : b.z;
    float y2 = inv ? fmaxf(b.w, 0.f) : b.w;
    x2 = fmaxf(x2, x1 + 1e-6f);
    y2 = fmaxf(y2, y1 + 1e-6f);
    return make_float4(x1, y1, x2, y2);
}

__global__ __launch_bounds__(NTHREADS)
void cutting_loss_main(const float4* __restrict__ logits,
                       const int*    __restrict__ labels,
                       const float4* __restrict__ pboxes,
                       const float4* __restrict__ tboxes,
                       const float*  __restrict__ clog,
                       const float*  __restrict__ ctgt,
                       float*        __restrict__ partials,
                       int n) {
    const int stride = gridDim.x * blockDim.x;
    float a_focal = 0.f, a_l1 = 0.f, a_giou = 0.f, a_bce = 0.f;

#pragma unroll 4
    for (int i = blockIdx.x * blockDim.x + threadIdx.x; i < n; i += stride) {
        // Speculative prefetch one grid-stride ahead (global_prefetch_b8).
        __builtin_prefetch((const void*)(logits + i + stride), 0, 0);
        __builtin_prefetch((const void*)(pboxes + i + stride), 0, 0);
        __builtin_prefetch((const void*)(tboxes + i + stride), 0, 0);

        // ---- focal loss (log_softmax over 4 classes) ----
        const float4 l = logits[i];
        const int lab  = labels[i];
        const float m  = fmaxf(fmaxf(l.x, l.y), fmaxf(l.z, l.w));
        const float es = __expf(l.x - m) + __expf(l.y - m) +
                         __expf(l.z - m) + __expf(l.w - m);
        const float lse = m + __logf(es);
        const float ll  = (lab == 0) ? l.x : (lab == 1) ? l.y : (lab == 2) ? l.z : l.w;
        const float ce  = lse - ll;            // -log p_label
        const float pt  = __expf(-ce);
        const float om  = 1.0f - pt;
        a_focal += om * om * ce;               // ALPHA=1, GAMMA=2

        // ---- L1 bbox ----
        const float4 pb = pboxes[i];
        const float4 tb = tboxes[i];
        a_l1 += fabsf(pb.x - tb.x) + fabsf(pb.y - tb.y) +
                fabsf(pb.z - tb.z) + fabsf(pb.w - tb.w);

        // ---- GIoU ----
        const float4 b1 = fix_box(pb);
        const float4 b2 = fix_box(tb);
        const float area1 = (b1.z - b1.x) * (b1.w - b1.y);
        const float area2 = (b2.z - b2.x) * (b2.w - b2.y);
        const float ltx = fmaxf(b1.x, b2.x), lty = fmaxf(b1.y, b2.y);
        const float rbx = fminf(b1.z, b2.z), rby = fminf(b1.w, b2.w);
        const float iw = fmaxf(rbx - ltx, 0.f), ih = fmaxf(rby - lty, 0.f);
        const float inter = iw * ih;
        const float uni   = area1 + area2 - inter;
        const float iou   = inter / (uni + 1e-7f);
        const float ex1 = fminf(b1.x, b2.x), ey1 = fminf(b1.y, b2.y);
        const float ex2 = fmaxf(b1.z, b2.z), ey2 = fmaxf(b1.w, b2.w);
        const float ew = fmaxf(ex2 - ex1, 0.f), eh = fmaxf(ey2 - ey1, 0.f);
        const float enc = ew * eh;
        const float giou = iou - (enc - uni) / (enc + 1e-7f);
        a_giou += fmaxf(1.0f - giou, 0.f);

        // ---- weighted BCE (sigmoid(x)<0.5 <=> x<0) ----
        // log1p(exp(-|x|)) == log(1 + exp(-|x|)); arg in (1,2] so the plain
        // log form is numerically safe and fully branchless (no libm paths).
        const float x = clog[i];
        const float t = ctgt[i];
        const float bce = fmaxf(x, 0.f) - x * t + __logf(1.0f + __expf(-fabsf(x)));
        const float w   = ((t == 1.0f) && (x < 0.0f)) ? 10.0f : 1.0f;
        a_bce += bce * w;
    }

    block_reduce4(a_focal, a_l1, a_giou, a_bce, &partials[(size_t)blockIdx.x * 4]);
}

__global__ __launch_bounds__(NTHREADS)
void cutting_loss_finalize(const float* __restrict__ partials,
                           float*       __restrict__ out,
                           int nrows, float inv_n) {
    float a0 = 0.f, a1 = 0.f, a2 = 0.f, a3 = 0.f;
    for (int r = threadIdx.x; r < nrows; r += blockDim.x) {
        const float4 p = reinterpret_cast<const float4*>(partials)[r];
        a0 += p.x; a1 += p.y; a2 += p.z; a3 += p.w;
    }
    __shared__ float res[4];
    block_reduce4(a0, a1, a2, a3, res);
    __syncthreads();
    if (threadIdx.x == 0) {
        const float lc  = fmaxf(res[0] * inv_n, 0.f);
        const float lb  = fmaxf(res[1] * inv_n * 0.25f, 0.f);  // mean over N*4
        const float lg  = fmaxf(res[2] * inv_n, 0.f);
        const float lcu = fmaxf(res[3] * inv_n, 0.f);
        out[0] = 1.0f * lc + 5.0f * lb + 2.0f * lg + 3.0f * lcu;
        out[1] = lc;
        out[2] = lb;
        out[3] = lg;
        out[4] = lcu;
    }
}

extern "C" void kernel_launch(void* const* d_in, const int* in_sizes, int n_in,
                              void* d_out, int out_size, void* d_ws, size_t ws_size,
                              hipStream_t stream) {
    const float4* logits = (const float4*)d_in[0];
    const int*    labels = (const int*)d_in[1];   // jax int64 -> int32 without x64
    const float4* pb     = (const float4*)d_in[2];
    const float4* tb     = (const float4*)d_in[3];
    const float*  cl     = (const float*)d_in[4];
    const float*  ct     = (const float*)d_in[5];
    float*        out    = (float*)d_out;
    float*        parts  = (float*)d_ws;          // NBLOCKS*4 floats = 32 KB
    const int n = in_sizes[1];                    // N

    cutting_loss_main<<<NBLOCKS, NTHREADS, 0, stream>>>(
        logits, labels, pb, tb, cl, ct, parts, n);
    cutting_loss_finalize<<<1, NTHREADS, 0, stream>>>(
        parts, out, NBLOCKS, 1.0f / (float)n);
}